// DetBenchTrain_120259084975
// MI455X (gfx1250) — compile-verified
//
#include <hip/hip_runtime.h>

#define A_TOTAL 49104
#define BATCH   8
#define MAX_GT  64
#define NCLS    90
#define NLVL    5
#define ALPHA   0.25f
#define GAMMA   1.5f
#define DELTA   0.1f
#define EPSF    1e-8f
#define NPART   4096
#define CLAIM_EMPTY 0x7FFFFFFF

typedef float v2f __attribute__((ext_vector_type(2)));
typedef float v4f __attribute__((ext_vector_type(4)));
typedef float v8f __attribute__((ext_vector_type(8)));

__device__ __constant__ int   c_feat[NLVL]   = {64, 32, 16, 8, 4};
__device__ __constant__ int   c_off[NLVL]    = {0, 36864, 46080, 48384, 48960};
__device__ __constant__ int   c_stride[NLVL] = {8, 16, 32, 64, 128};
__device__ __constant__ float c_oct[3] = {1.0f, 1.25992104989487316f, 1.58740105196819947f};

// ---------------- anchors ----------------
__global__ __launch_bounds__(256) void anchors_k(float* __restrict__ anchors) {
    int a = blockIdx.x * 256 + threadIdx.x;
    if (a >= A_TOTAL) return;
    int li = 0;
#pragma unroll
    for (int i = 1; i < NLVL; ++i) if (a >= c_off[i]) li = i;
    int rem = a - c_off[li];
    int f = c_feat[li];
    int stride = c_stride[li];
    int k = rem % 9;
    int loc = rem / 9;
    int x = loc % f, y = loc / f;
    int octv = k / 3, asp = k % 3;
    float base = 4.0f * (float)stride * c_oct[octv];
    float ax = (asp == 0) ? 1.0f : (asp == 1 ? 1.4f : 0.7f);
    float ay = (asp == 0) ? 1.0f : (asp == 1 ? 0.7f : 1.4f);
    float hh = base * ay * 0.5f, hw = base * ax * 0.5f;
    float cy = ((float)y + 0.5f) * (float)stride;
    float cx = ((float)x + 0.5f) * (float)stride;
    float4 box = make_float4(cy - hh, cx - hw, cy + hh, cx + hw);
    ((float4*)anchors)[a] = box;
}

// ---------------- init ----------------
__global__ __launch_bounds__(256) void init_k(int* claim, int* numpos, float* partials) {
    int gid = blockIdx.x * 256 + threadIdx.x;
    int tot = gridDim.x * 256;
    for (int i = gid; i < BATCH * A_TOTAL; i += tot) claim[i] = CLAIM_EMPTY;
    for (int i = gid; i < NPART; i += tot) partials[i] = 0.0f;
    if (gid == 0) numpos[0] = 0;
}

__device__ inline float iou_f(float g0, float g1, float g2, float g3,
                              float a0, float a1, float a2, float a3) {
    float yi1 = fmaxf(g0, a0), xi1 = fmaxf(g1, a1);
    float yi2 = fminf(g2, a2), xi2 = fminf(g3, a3);
    float inter = fmaxf(yi2 - yi1, 0.0f) * fmaxf(xi2 - xi1, 0.0f);
    float ag = (g2 - g0) * (g3 - g1);
    float aa = (a2 - a0) * (a3 - a1);
    return inter / (ag + aa - inter + EPSF);
}

// ---------------- per-anchor argmax over GT ----------------
__global__ __launch_bounds__(256) void match_anchor_k(const float* __restrict__ anchors,
                                                      const float* __restrict__ gt_boxes,
                                                      const int* __restrict__ gt_labels,
                                                      int* __restrict__ matches_pre) {
    __shared__ float gtb[MAX_GT * 4];
    __shared__ int gval[MAX_GT];
    int b = blockIdx.y;
    int t = threadIdx.x;
    gtb[t] = gt_boxes[b * (MAX_GT * 4) + t];
    if (t < MAX_GT) gval[t] = (gt_labels[b * MAX_GT + t] > 0) ? 1 : 0;
    __syncthreads();
    int a = blockIdx.x * 256 + t;
    if (a >= A_TOTAL) return;
    float4 an = ((const float4*)anchors)[a];
    float best = -2.0f; int bg = 0;
#pragma unroll 4
    for (int g = 0; g < MAX_GT; ++g) {
        float v = gval[g] ? iou_f(gtb[4*g], gtb[4*g+1], gtb[4*g+2], gtb[4*g+3],
                                  an.x, an.y, an.z, an.w)
                          : -1.0f;
        if (v > best) { best = v; bg = g; }   // strict > : first-occurrence argmax
    }
    matches_pre[b * A_TOTAL + a] = (best >= 0.5f) ? bg : -1;
}

// ---------------- per-GT argmax over anchors -> claim ----------------
__global__ __launch_bounds__(256) void gt_best_k(const float* __restrict__ anchors,
                                                 const float* __restrict__ gt_boxes,
                                                 const int* __restrict__ gt_labels,
                                                 int* __restrict__ claim) {
    __shared__ unsigned long long red[256];
    int bg = blockIdx.x;
    int b = bg >> 6, g = bg & 63;
    if (gt_labels[b * MAX_GT + g] <= 0) return;  // uniform over block
    const float* G = gt_boxes + (b * MAX_GT + g) * 4;
    float g0 = G[0], g1 = G[1], g2 = G[2], g3 = G[3];
    unsigned long long best = 0ULL;
    for (int a = threadIdx.x; a < A_TOTAL; a += 256) {
        float4 an = ((const float4*)anchors)[a];
        float v = iou_f(g0, g1, g2, g3, an.x, an.y, an.z, an.w);  // v >= 0 always
        unsigned long long key = ((unsigned long long)__float_as_uint(v) << 32)
                               | (unsigned long long)(0xFFFFFFFFu - (unsigned)a);
        if (key > best) best = key;  // max iou, ties -> lowest anchor index
    }
    red[threadIdx.x] = best;
    __syncthreads();
    for (int s = 128; s > 0; s >>= 1) {
        if (threadIdx.x < s) { if (red[threadIdx.x + s] > red[threadIdx.x]) red[threadIdx.x] = red[threadIdx.x + s]; }
        __syncthreads();
    }
    if (threadIdx.x == 0) {
        int a = (int)(0xFFFFFFFFu - (unsigned)(red[0] & 0xFFFFFFFFu));
        atomicMin(&claim[b * A_TOTAL + a], g);   // lowest g wins ties (argmax(0) semantics)
    }
}

// ---------------- finalize: override matches, cls targets, pos count ----------------
__global__ __launch_bounds__(256) void finalize_k(const int* __restrict__ claim,
                                                  const int* __restrict__ matches_pre,
                                                  const int* __restrict__ gt_labels,
                                                  int* __restrict__ matches,
                                                  int* __restrict__ cls_t,
                                                  int* __restrict__ numpos) {
    __shared__ int red[256];
    int i = blockIdx.x * 256 + threadIdx.x;
    int cnt = 0;
    if (i < BATCH * A_TOTAL) {
        int b = i / A_TOTAL;
        int c = claim[i];
        int m = (c != CLAIM_EMPTY) ? c : matches_pre[i];
        matches[i] = m;
        cls_t[i] = (m >= 0) ? (gt_labels[b * MAX_GT + m] - 1) : -1;
        cnt = (m >= 0) ? 1 : 0;
    }
    red[threadIdx.x] = cnt;
    __syncthreads();
    for (int s = 128; s > 0; s >>= 1) {
        if (threadIdx.x < s) red[threadIdx.x] += red[threadIdx.x + s];
        __syncthreads();
    }
    if (threadIdx.x == 0 && red[0] > 0) atomicAdd(numpos, red[0]);
}

__device__ inline float block_sum_f(float v) {
    __shared__ float sm[256];
    sm[threadIdx.x] = v;
    __syncthreads();
    for (int s = 128; s > 0; s >>= 1) {
        if (threadIdx.x < s) sm[threadIdx.x] += sm[threadIdx.x + s];
        __syncthreads();
    }
    return sm[0];
}

// ---------------- focal loss streaming kernel (templated on level geometry) ----------------
// f is a power of two -> all /f, %f are shifts; decode hoisted to once per float4
// (all 4 lanes of an aligned quad share b, ch, y since f >= 4).
template <int F, int LVL_OFF>
__global__ __launch_bounds__(256) void focal_k(const float* __restrict__ logits,
                                               const int* __restrict__ cls_t,
                                               float* __restrict__ partials,
                                               int N4) {
    float acc = 0.0f;
    int tot = gridDim.x * 256;
    for (int i = blockIdx.x * 256 + threadIdx.x; i < N4; i += tot) {
        v4f v = __builtin_nontemporal_load(((const v4f*)logits) + i);  // read-once stream
        int e0 = i * 4;
        int x0 = e0 % F;  int t1 = e0 / F;
        int y  = t1 % F;  t1 /= F;
        int ch = t1 % (9 * NCLS);
        int b  = t1 / (9 * NCLS);
        int k  = ch / NCLS;
        int c  = ch - k * NCLS;
        int a0 = LVL_OFF + (y * F + x0) * 9 + k;
        const int* ctp = cls_t + b * A_TOTAL + a0;
#pragma unroll
        for (int j = 0; j < 4; ++j) {
            float l = v[j];
            int ct = ctp[9 * j];
            bool oh = (c == ct);
            float absl = fabsf(l);
            float sp_nabs = __logf(1.0f + __expf(-absl));     // softplus(-|l|)
            float lo = oh ? l : 0.0f;
            float ce = fmaxf(l, 0.0f) - lo + sp_nabs;
            float sp_nl = sp_nabs + fmaxf(-l, 0.0f);          // softplus(-l), stable
            float mod = __expf(-GAMMA * (lo + sp_nl));
            float w = oh ? ALPHA : (1.0f - ALPHA);
            acc += w * mod * ce;
        }
    }
    float bsum = block_sum_f(acc);
    if (threadIdx.x == 0) partials[blockIdx.x] = bsum;
}

// ---------------- huber box loss streaming kernel (templated) ----------------
template <int F, int LVL_OFF>
__global__ __launch_bounds__(256) void box_k(const float* __restrict__ bp,
                                             const int* __restrict__ matches,
                                             const float* __restrict__ gt_boxes,
                                             const float* __restrict__ anchors,
                                             float* __restrict__ partials,
                                             int N4) {
    float acc = 0.0f;
    int tot = gridDim.x * 256;
    for (int i = blockIdx.x * 256 + threadIdx.x; i < N4; i += tot) {
        v4f v = __builtin_nontemporal_load(((const v4f*)bp) + i);
        int e0 = i * 4;
        int x0 = e0 % F;  int t1 = e0 / F;
        int y  = t1 % F;  t1 /= F;
        int ch = t1 % 36;
        int b  = t1 / 36;
        int k  = ch >> 2;
        int d  = ch & 3;
        int a0 = LVL_OFF + (y * F + x0) * 9 + k;
#pragma unroll
        for (int j = 0; j < 4; ++j) {
            int a = a0 + 9 * j;
            int m = matches[b * A_TOTAL + a];
            if (m >= 0) {
                const float* G = gt_boxes + (b * MAX_GT + m) * 4;
                float4 an = ((const float4*)anchors)[a];
                float ha = an.z - an.x + EPSF;
                float wa = an.w - an.y + EPSF;
                float td;
                if (d == 0)      td = ((G[0] + G[2]) - (an.x + an.z)) * 0.5f / ha;
                else if (d == 1) td = ((G[1] + G[3]) - (an.y + an.w)) * 0.5f / wa;
                else if (d == 2) td = __logf((G[2] - G[0] + EPSF) / ha);
                else             td = __logf((G[3] - G[1] + EPSF) / wa);
                if (td != 0.0f) {
                    float ae = fabsf(v[j] - td);
                    float q = fminf(ae, DELTA);
                    acc += 0.5f * q * q + DELTA * (ae - q);
                }
            }
        }
    }
    float bsum = block_sum_f(acc);
    if (threadIdx.x == 0) partials[blockIdx.x] = bsum * 12.5f;  // BOX_LOSS_W / 4
}

// ---------------- final deterministic reduction via V_WMMA_F32_16X16X4_F32 ----------------
// A = all-ones 16x4, B = 64 partials/iteration. Sum of all D entries == 16 * sum(B inputs),
// independent of the fragment->matrix element mapping (bijective).
__global__ __launch_bounds__(32) void reduce_k(const float* __restrict__ partials,
                                               const int* __restrict__ numpos,
                                               float* __restrict__ out) {
    __shared__ float red[32];
    int lane = threadIdx.x;
    v8f c = {};
    v2f ones; ones[0] = 1.0f; ones[1] = 1.0f;
    for (int i = 0; i < NPART; i += 64) {
        v2f bf;
        bf[0] = partials[i + lane * 2];
        bf[1] = partials[i + lane * 2 + 1];
        c = __builtin_amdgcn_wmma_f32_16x16x4_f32(false, ones, false, bf,
                                                  (short)0, c, false, false);
    }
    float s = 0.0f;
#pragma unroll
    for (int r = 0; r < 8; ++r) s += c[r];
    red[lane] = s;
    __syncthreads();
    if (lane == 0) {
        float totv = 0.0f;
        for (int i = 0; i < 32; ++i) totv += red[i];
        totv *= (1.0f / 16.0f);
        float nps = (float)numpos[0] + 1.0f;
        out[0] = totv / nps;
    }
}

// ---------------- host launcher ----------------
extern "C" void kernel_launch(void* const* d_in, const int* in_sizes, int n_in,
                              void* d_out, int out_size, void* d_ws, size_t ws_size,
                              hipStream_t stream) {
    const float* cls_in[NLVL];
    const float* box_in[NLVL];
    for (int i = 0; i < NLVL; ++i) {
        cls_in[i] = (const float*)d_in[i];
        box_in[i] = (const float*)d_in[5 + i];
    }
    const float* gt_boxes = (const float*)d_in[10];
    const int*   gt_labels = (const int*)d_in[11];

    // workspace layout
    float* anchors     = (float*)d_ws;                       // A_TOTAL*4 floats
    int*   matches_pre = (int*)(anchors + A_TOTAL * 4);      // B*A
    int*   claim       = matches_pre + BATCH * A_TOTAL;      // B*A
    int*   matches     = claim + BATCH * A_TOTAL;            // B*A
    int*   cls_t       = matches + BATCH * A_TOTAL;          // B*A
    int*   numpos      = cls_t + BATCH * A_TOTAL;            // 1 (+3 pad)
    float* partials    = (float*)(numpos + 4);               // NPART floats

    anchors_k<<<(A_TOTAL + 255) / 256, 256, 0, stream>>>(anchors);
    init_k<<<1536, 256, 0, stream>>>(claim, numpos, partials);
    match_anchor_k<<<dim3((A_TOTAL + 255) / 256, BATCH), 256, 0, stream>>>(
        anchors, gt_boxes, gt_labels, matches_pre);
    gt_best_k<<<BATCH * MAX_GT, 256, 0, stream>>>(anchors, gt_boxes, gt_labels, claim);
    finalize_k<<<(BATCH * A_TOTAL + 255) / 256, 256, 0, stream>>>(
        claim, matches_pre, gt_labels, matches, cls_t, numpos);

    // per-level streaming losses (templated geometry; fixed partial slots)
    {
        const int ncls4_0 = (BATCH * 9 * NCLS * 64 * 64) / 4;
        const int ncls4_1 = (BATCH * 9 * NCLS * 32 * 32) / 4;
        const int ncls4_2 = (BATCH * 9 * NCLS * 16 * 16) / 4;
        const int ncls4_3 = (BATCH * 9 * NCLS * 8 * 8) / 4;
        const int ncls4_4 = (BATCH * 9 * NCLS * 4 * 4) / 4;
        focal_k<64, 0>    <<<1024, 256, 0, stream>>>(cls_in[0], cls_t, partials + 0,    ncls4_0);
        focal_k<32, 36864><<< 512, 256, 0, stream>>>(cls_in[1], cls_t, partials + 1024, ncls4_1);
        focal_k<16, 46080><<< 256, 256, 0, stream>>>(cls_in[2], cls_t, partials + 1536, ncls4_2);
        focal_k< 8, 48384><<< 128, 256, 0, stream>>>(cls_in[3], cls_t, partials + 1792, ncls4_3);
        focal_k< 4, 48960><<<  64, 256, 0, stream>>>(cls_in[4], cls_t, partials + 1920, ncls4_4);

        const int nbox4_0 = (BATCH * 36 * 64 * 64) / 4;
        const int nbox4_1 = (BATCH * 36 * 32 * 32) / 4;
        const int nbox4_2 = (BATCH * 36 * 16 * 16) / 4;
        const int nbox4_3 = (BATCH * 36 * 8 * 8) / 4;
        const int nbox4_4 = (BATCH * 36 * 4 * 4) / 4;
        box_k<64, 0>    <<<256, 256, 0, stream>>>(box_in[0], matches, gt_boxes, anchors, partials + 1984, nbox4_0);
        box_k<32, 36864><<<128, 256, 0, stream>>>(box_in[1], matches, gt_boxes, anchors, partials + 2240, nbox4_1);
        box_k<16, 46080><<< 64, 256, 0, stream>>>(box_in[2], matches, gt_boxes, anchors, partials + 2368, nbox4_2);
        box_k< 8, 48384><<< 32, 256, 0, stream>>>(box_in[3], matches, gt_boxes, anchors, partials + 2432, nbox4_3);
        box_k< 4, 48960><<< 16, 256, 0, stream>>>(box_in[4], matches, gt_boxes, anchors, partials + 2464, nbox4_4);
    }

    reduce_k<<<1, 32, 0, stream>>>(partials, numpos, (float*)d_out);
    (void)in_sizes; (void)n_in; (void)out_size; (void)ws_size;
}